// CRF_8581344657961
// MI455X (gfx1250) — compile-verified
//
#include <hip/hip_runtime.h>
#include <hip/hip_bf16.h>

// CRF forward-backward + Viterbi for K=4, Tp1=513, Y=256 on gfx1250.
//   phase1: per-chunk exp-space matrix chain products via V_WMMA_F32_16X16X4_F32
//           A panels staged by TDM (tensor_load_to_lds, double-buffered, LDS pad),
//           B panels exp'd+scaled once into LDS; 8 accumulator tiles per wave.
//   phase2: sequential prefix over 129 chunk products (shuffle reductions)
//   phase3: parallel intra-chunk log-sum-exp vector scans
//   p12 elementwise, Viterbi max-plus scan + LDS backtrace.

#define NEGV  (-1.0e30f)
#define YDIM  256
#define YY    65536
#define TP1   513
#define TLEN  512
#define CHUNK 4
#define NCH   129   // 128 chunks of 4 + 1 chunk of 1
#define PAD   20    // LDS row pad: even (v2f aligned), 16B-aligned rows (TDM B128),
                    // stride-20 => 64 distinct banks per wave fragment read

typedef float v2f __attribute__((ext_vector_type(2)));
typedef float v8f __attribute__((ext_vector_type(8)));
typedef unsigned int v4u __attribute__((ext_vector_type(4)));
typedef int v8i __attribute__((ext_vector_type(8)));
typedef int v4i __attribute__((ext_vector_type(4)));

// ---------------------------------------------------------------- logM
__global__ void k_logM(const float* __restrict__ f, const float* __restrict__ w,
                       float* __restrict__ logM) {
    const float w0 = w[0], w1 = w[1], w2 = w[2], w3 = w[3];
    const size_t PLANE = (size_t)TP1 * YY;
    const size_t total4 = PLANE / 4;
    for (size_t n4 = (size_t)blockIdx.x * blockDim.x + threadIdx.x; n4 < total4;
         n4 += (size_t)gridDim.x * blockDim.x) {
        size_t n = n4 * 4;
        float4 a = *(const float4*)(f + n);
        float4 b = *(const float4*)(f + PLANE + n);
        float4 c = *(const float4*)(f + 2 * PLANE + n);
        float4 d = *(const float4*)(f + 3 * PLANE + n);
        float4 o;
        o.x = w0 * a.x + w1 * b.x + w2 * c.x + w3 * d.x;
        o.y = w0 * a.y + w1 * b.y + w2 * c.y + w3 * d.y;
        o.z = w0 * a.z + w1 * b.z + w2 * c.z + w3 * d.z;
        o.w = w0 * a.w + w1 * b.w + w2 * c.w + w3 * d.w;
        int t = (int)(n >> 16);
        int i = (int)(n >> 8) & 255;
        int j = (int)n & 255;   // components cover j..j+3, same t,i
        if (t == 0 && i != 0) { o.x = NEGV; o.y = NEGV; o.z = NEGV; o.w = NEGV; }
        if (t == TLEN) {        // keep only column 0
            if (j != 0) o.x = NEGV;
            o.y = NEGV; o.z = NEGV; o.w = NEGV;
        }
        *(float4*)(logM + n) = o;
    }
}

// TDM: DMA a 256-row x 16-col fp32 panel (row stride 256 elems) from global into
// LDS at lds_byte, inserting 4 pad DWORDs after every 16 DWORDs (=> PAD=20 rows).
__device__ __forceinline__ void tdm_load_panel(const float* gsrc, unsigned lds_byte) {
    unsigned long long ga = (unsigned long long)(uintptr_t)gsrc;
    v4u g0;
    g0.x = 1u;                                            // count=1 (valid user D#)
    g0.y = lds_byte;                                      // lds_addr
    g0.z = (unsigned)(ga & 0xFFFFFFFFu);                  // global_addr[31:0]
    g0.w = (unsigned)((ga >> 32) & 0x01FFFFFFu) | (2u << 30);  // addr[56:32] | type=2
    v8i g1;
    g1[0] = (2 << 16) | (1 << 20) | (3 << 22) | (3 << 25);
            // data_size=4B, pad_enable, pad_interval=16DW, pad_amount=4DW
    g1[1] = (int)(256u << 16);                            // tensor_dim0 = 256
    g1[2] = (int)(256u << 16);                            // tensor_dim1 = 256
    g1[3] = (int)(16u << 16);                             // tile_dim0 = 16
    g1[4] = 256;                                          // tile_dim1 = 256
    g1[5] = 256;                                          // tensor_dim0_stride = 256
    g1[6] = 0;
    g1[7] = 0;
    v4i gz4 = {0, 0, 0, 0};
    v8i gz8 = {0, 0, 0, 0, 0, 0, 0, 0};
    __builtin_amdgcn_tensor_load_to_lds(g0, g1, gz4, gz4, gz8, 0);
}

// ------------------------------------------------- phase1: chunk products (WMMA)
// R <- R @ (inv*exp(logM[t])) repeated over chunk; scalar rescale per step folded
// into the B panel so the A panel is a raw copy (TDM-friendly).
__global__ __launch_bounds__(1024) void k_chunk_prod(const float* __restrict__ logM,
                                                     float* __restrict__ Rbuf,
                                                     float* __restrict__ logS) {
    __shared__ float As[2][256 * PAD];   // double-buffered A panels (TDM dest)
    __shared__ float Bs[256 * PAD];      // B panel: 256 cols x 16 k (inv*exp)
    __shared__ int   s_maxbits;
    __shared__ float s_inv, s_logS;
    const int c    = blockIdx.x;
    const int t0   = c * CHUNK;
    const int L    = min(CHUNK, TP1 - t0);
    const int tid  = threadIdx.x;
    const int wave = tid >> 5, lane = tid & 31;
    const int rbl  = wave >> 4;           // 0..1: 16-row sub-block within pair
    const int cb   = wave & 15;           // 16-col tile
    const int l15  = lane & 15;
    const int ksub = (lane >> 4) << 1;    // 0 or 2 (always even)
    const unsigned ldsA0 = (unsigned)(uintptr_t)&As[0][0];
    const unsigned ldsA1 = (unsigned)(uintptr_t)&As[1][0];
    float* buf0 = Rbuf + (size_t)c * 2 * YY;
    float* buf1 = buf0 + YY;

    // init R = exp(logM[t0])
    const float* M0 = logM + (size_t)t0 * YY;
    for (int v = tid; v < YY; v += 1024) buf0[v] = __expf(M0[v]);
    if (tid == 0) { s_maxbits = __float_as_int(1.0f); s_logS = 0.f; }
    __threadfence();
    __syncthreads();

    int cur = 0;
    for (int s = 1; s < L; ++s) {
        const float* Mt = logM + (size_t)(t0 + s) * YY;
        float* Rc = cur ? buf1 : buf0;
        float* Rn = cur ? buf0 : buf1;
        if (tid == 0) {                       // apply pending rescale
            float m = __int_as_float(s_maxbits);
            s_inv   = 1.0f / m;
            s_logS += __logf(m);
            s_maxbits = __float_as_int(0.0f);
        }
        __syncthreads();
        const float inv = s_inv;
        const v8f zero = {0.f, 0.f, 0.f, 0.f, 0.f, 0.f, 0.f, 0.f};
        v8f acc[8];
#pragma unroll
        for (int h = 0; h < 8; ++h) acc[h] = zero;

        if (wave == 0) tdm_load_panel(Rc, ldsA0);   // prefetch A panel 0

        for (int kp = 0; kp < 16; ++kp) {
            const int kbase = kp * 16;
            // stage scaled+exp'd B panel transposed: Bs[col][k] (coalesced reads)
            for (int v = tid; v < 4096; v += 1024) {
                int kk = v >> 8, col = v & 255;
                Bs[col * PAD + kk] = inv * __expf(Mt[(kbase + kk) * 256 + col]);
            }
            if (wave == 0) __builtin_amdgcn_s_wait_tensorcnt(0);  // A panel kp ready
            __syncthreads();
            if (wave == 0 && kp < 15)                 // DMA next A panel during WMMAs
                tdm_load_panel(Rc + (kp + 1) * 16, (kp & 1) ? ldsA0 : ldsA1);
            const float* Ap = As[kp & 1];
#pragma unroll
            for (int ks = 0; ks < 4; ++ks) {
                const int kl = ks * 4 + ksub;                     // even
                v2f b = *(const v2f*)&Bs[(cb * 16 + l15) * PAD + kl];
#pragma unroll
                for (int h = 0; h < 8; ++h) {
                    v2f a = *(const v2f*)&Ap[(h * 32 + rbl * 16 + l15) * PAD + kl];
                    acc[h] = __builtin_amdgcn_wmma_f32_16x16x4_f32(
                                 false, a, false, b, (short)0, acc[h], false, false);
                }
            }
            __syncthreads();
        }
        // write result tiles + block max
        float wmax = 0.f;
        const int ocol = cb * 16 + l15;
        const int rsub = (lane >> 4) << 3;
#pragma unroll
        for (int h = 0; h < 8; ++h) {
            const int orow0 = h * 32 + rbl * 16 + rsub;
#pragma unroll
            for (int r = 0; r < 8; ++r) {
                float vv = acc[h][r];
                Rn[(orow0 + r) * 256 + ocol] = vv;
                wmax = fmaxf(wmax, vv);
            }
        }
#pragma unroll
        for (int off = 16; off; off >>= 1)
            wmax = fmaxf(wmax, __shfl_xor(wmax, off, 32));
        if (lane == 0) atomicMax(&s_maxbits, __float_as_int(wmax));
        __threadfence();                 // make Rn visible across both CUs / TDM
        cur ^= 1;
        __syncthreads();
    }
    if (tid == 0) logS[c] = s_logS;   // scale NOT including final step's max
}

// ------------------------------------- phase2: sequential prefix over chunks
__global__ void k_boundary(const float* __restrict__ Rbuf, const float* __restrict__ logS,
                           float* __restrict__ Abound, float* __restrict__ Bbound,
                           float* __restrict__ logZp) {
    __shared__ float v[256], nv[256];
    __shared__ float wred[8];
    __shared__ float s_shift, s_m;
    const int j = threadIdx.x;
    const int lane = j & 31, wv = j >> 5;
    // ---- forward: alpha boundaries at each chunk start
    v[j] = (j == 0) ? 1.f : 0.f;
    if (j == 0) s_shift = 0.f;
    __syncthreads();
    for (int c = 0; c < NCH; ++c) {
        Abound[c * 256 + j] = (v[j] > 0.f) ? __logf(v[j]) + s_shift : NEGV;
        const int L = min(CHUNK, TP1 - c * CHUNK);
        const float* P = Rbuf + (size_t)c * 2 * YY + (size_t)((L - 1) & 1) * YY;
        float acc = 0.f;
        for (int i = 0; i < 256; ++i) acc = __fmaf_rn(v[i], P[i * 256 + j], acc);
        nv[j] = acc;
        float x = acc;
#pragma unroll
        for (int off = 16; off; off >>= 1) x = fmaxf(x, __shfl_xor(x, off, 32));
        if (lane == 0) wred[wv] = x;
        __syncthreads();
        if (j == 0) {
            float m = wred[0];
#pragma unroll
            for (int p = 1; p < 8; ++p) m = fmaxf(m, wred[p]);
            s_m = (m > 0.f) ? m : 1.f;
            s_shift += logS[c] + __logf(s_m);
        }
        __syncthreads();
        v[j] = nv[j] / s_m;
        __syncthreads();
    }
    {   // logZ = log(sum v) + shift
        float x = v[j];
#pragma unroll
        for (int off = 16; off; off >>= 1) x += __shfl_xor(x, off, 32);
        if (lane == 0) wred[wv] = x;
        __syncthreads();
        if (j == 0) {
            float ssum = 0.f;
#pragma unroll
            for (int p = 0; p < 8; ++p) ssum += wred[p];
            *logZp = __logf(ssum) + s_shift;
        }
        __syncthreads();
    }
    // ---- backward: beta boundaries at each chunk end
    v[j] = (j == 0) ? 1.f : 0.f;
    if (j == 0) s_shift = 0.f;
    __syncthreads();
    for (int c = NCH - 1; c >= 0; --c) {
        Bbound[(c + 1) * 256 + j] = (v[j] > 0.f) ? __logf(v[j]) + s_shift : NEGV;
        const int L = min(CHUNK, TP1 - c * CHUNK);
        const float* P = Rbuf + (size_t)c * 2 * YY + (size_t)((L - 1) & 1) * YY;
        float acc = 0.f;
        for (int q = 0; q < 256; ++q) acc = __fmaf_rn(P[j * 256 + q], v[q], acc);
        nv[j] = acc;
        float x = acc;
#pragma unroll
        for (int off = 16; off; off >>= 1) x = fmaxf(x, __shfl_xor(x, off, 32));
        if (lane == 0) wred[wv] = x;
        __syncthreads();
        if (j == 0) {
            float m = wred[0];
#pragma unroll
            for (int p = 1; p < 8; ++p) m = fmaxf(m, wred[p]);
            s_m = (m > 0.f) ? m : 1.f;
            s_shift += logS[c] + __logf(s_m);
        }
        __syncthreads();
        v[j] = nv[j] / s_m;
        __syncthreads();
    }
    Bbound[j] = (v[j] > 0.f) ? __logf(v[j]) + s_shift : NEGV;
}

// -------------------------------- phase3 fwd: intra-chunk log-sum-exp scan
__global__ void k_fwd(const float* __restrict__ logM, const float* __restrict__ Abound,
                      float* __restrict__ LA) {
    __shared__ float la[256];
    const int c = blockIdx.x, j = threadIdx.x;
    const int t0 = c * CHUNK;
    const int L  = min(CHUNK, TP1 - t0);
    la[j] = Abound[c * 256 + j];
    __syncthreads();
    LA[(size_t)t0 * 256 + j] = la[j];
    for (int s = 1; s < L; ++s) {
        const float* Mt = logM + (size_t)(t0 + s - 1) * YY;
        float m = -3.0e38f, ssum = 0.f;
        for (int i = 0; i < 256; ++i) {
            float x  = la[i] + Mt[i * 256 + j];
            float nm = fmaxf(m, x);
            ssum = ssum * __expf(m - nm) + __expf(x - nm);
            m = nm;
        }
        float res = m + __logf(ssum);
        __syncthreads();
        la[j] = res;
        __syncthreads();
        LA[(size_t)(t0 + s) * 256 + j] = res;
    }
}

// -------------------------------- phase3 bwd: wave-per-row log-sum-exp scan
__global__ void k_bwd(const float* __restrict__ logM, const float* __restrict__ Bbound,
                      float* __restrict__ LB) {
    __shared__ float lb[256], lbn[256];
    const int c = blockIdx.x;
    const int tid = threadIdx.x, wave = tid >> 5, lane = tid & 31;
    const int t0 = c * CHUNK;
    const int L  = min(CHUNK, TP1 - t0);
    const int tend = t0 + L;
    if (tid < 256) lb[tid] = Bbound[(c + 1) * 256 + tid];
    __syncthreads();
    if (tid < 256) LB[(size_t)(tend - 1) * 256 + tid] = lb[tid];   // beta[tend]
    for (int t = tend - 1; t > t0; --t) {
        const float* Mt = logM + (size_t)t * YY;
        for (int row = wave; row < 256; row += 8) {
            float m = -3.0e38f, ssum = 0.f;
            for (int jj = lane; jj < 256; jj += 32) {
                float x  = Mt[row * 256 + jj] + lb[jj];
                float nm = fmaxf(m, x);
                ssum = ssum * __expf(m - nm) + __expf(x - nm);
                m = nm;
            }
#pragma unroll
            for (int off = 16; off; off >>= 1) {
                float om = __shfl_xor(m, off, 32);
                float os = __shfl_xor(ssum, off, 32);
                float nm = fmaxf(m, om);
                ssum = ssum * __expf(m - nm) + os * __expf(om - nm);
                m = nm;
            }
            if (lane == 0) lbn[row] = m + __logf(ssum);
        }
        __syncthreads();
        if (tid < 256) lb[tid] = lbn[tid];
        __syncthreads();
        if (tid < 256) LB[(size_t)(t - 1) * 256 + tid] = lb[tid];  // beta[t]
    }
}

// ---------------------------------------------------------------- p12
__global__ void k_p12(const float* __restrict__ logM, const float* __restrict__ LA,
                      const float* __restrict__ LB, const float* __restrict__ logZp,
                      float* __restrict__ out) {
    const float lz = *logZp;
    const size_t total4 = (size_t)TP1 * YY / 4;
    for (size_t n4 = (size_t)blockIdx.x * blockDim.x + threadIdx.x; n4 < total4;
         n4 += (size_t)gridDim.x * blockDim.x) {
        size_t n = n4 * 4;
        int t = (int)(n >> 16);
        int i = (int)(n >> 8) & 255;
        int j = (int)n & 255;
        float4 m4 = *(const float4*)(logM + n);
        float base = LA[t * 256 + i] - lz;
        float4 o;
        o.x = __expf(base + m4.x + LB[t * 256 + j + 0]);
        o.y = __expf(base + m4.y + LB[t * 256 + j + 1]);
        o.z = __expf(base + m4.z + LB[t * 256 + j + 2]);
        o.w = __expf(base + m4.w + LB[t * 256 + j + 3]);
        *(float4*)(out + n) = o;
    }
}

// ---------------------------------------------------------------- Viterbi scan
__global__ __launch_bounds__(1024) void k_vit(const float* __restrict__ logM,
                                              unsigned char* __restrict__ vidx,
                                              int* __restrict__ vlast) {
    __shared__ float delta[256];
    __shared__ float pmax[1024];
    __shared__ int   parg[1024];
    const int tid = threadIdx.x;
    const int q = tid >> 8, j = tid & 255;
    if (tid < 256) delta[tid] = logM[tid];        // logM[0][0][:]
    __syncthreads();
    for (int t = 1; t < TLEN; ++t) {
        const float* Mt = logM + (size_t)t * YY;
        float bm = -3.0e38f; int ba = 0;
        const int i0 = q * 64;
        for (int i = i0; i < i0 + 64; ++i) {
            float x = delta[i] + Mt[i * 256 + j];
            if (x > bm) { bm = x; ba = i; }
        }
        pmax[tid] = bm; parg[tid] = ba;
        __syncthreads();
        if (q == 0) {
            float best = pmax[j]; int bi = parg[j];
            for (int p = 1; p < 4; ++p) {
                float v2 = pmax[p * 256 + j];
                if (v2 > best) { best = v2; bi = parg[p * 256 + j]; }
            }
            vidx[(size_t)(t - 1) * 256 + j] = (unsigned char)bi;
            delta[j] = best;
        }
        __syncthreads();
    }
    if (tid == 0) {
        float best = delta[0]; int bi = 0;
        for (int i = 1; i < 256; ++i)
            if (delta[i] > best) { best = delta[i]; bi = i; }
        *vlast = bi;
    }
}

// ---------------------------------------------------------------- backtrace
__global__ __launch_bounds__(1024) void k_back(const unsigned char* __restrict__ vidx,
                                               const int* __restrict__ vlast,
                                               float* __restrict__ out_path) {
    extern __shared__ unsigned char sidx[];        // 511*256 bytes dynamic LDS
    for (int v = threadIdx.x; v < (TLEN - 1) * 256; v += 1024) sidx[v] = vidx[v];
    __syncthreads();
    if (threadIdx.x == 0) {
        int y = *vlast;
        out_path[TLEN - 1] = (float)y;
        for (int s = TLEN - 2; s >= 0; --s) {
            y = sidx[s * 256 + y];
            out_path[s] = (float)y;
        }
    }
}

// ---------------------------------------------------------------- launch
extern "C" void kernel_launch(void* const* d_in, const int* in_sizes, int n_in,
                              void* d_out, int out_size, void* d_ws, size_t ws_size,
                              hipStream_t stream) {
    (void)in_sizes; (void)n_in; (void)out_size; (void)ws_size;
    const float* f = (const float*)d_in[0];
    const float* w = (const float*)d_in[1];
    float* out = (float*)d_out;

    const size_t NLOGM = (size_t)TP1 * YY;
    float* logM   = (float*)d_ws;
    float* Rbuf   = logM + NLOGM;
    float* logS   = Rbuf + (size_t)NCH * 2 * YY;
    float* Abound = logS + NCH;
    float* Bbound = Abound + NCH * 256;
    float* LA     = Bbound + (NCH + 1) * 256;
    float* LB     = LA + (size_t)TP1 * 256;
    float* logZp  = LB + (size_t)TP1 * 256;
    int*   vlast  = (int*)(logZp + 1);
    unsigned char* vidx = (unsigned char*)(vlast + 1);

    k_logM<<<2048, 256, 0, stream>>>(f, w, logM);
    k_chunk_prod<<<NCH, 1024, 0, stream>>>(logM, Rbuf, logS);
    k_boundary<<<1, 256, 0, stream>>>(Rbuf, logS, Abound, Bbound, logZp);
    k_fwd<<<NCH, 256, 0, stream>>>(logM, Abound, LA);
    k_bwd<<<NCH, 256, 0, stream>>>(logM, Bbound, LB);
    k_p12<<<2048, 256, 0, stream>>>(logM, LA, LB, logZp, out);
    k_vit<<<1, 1024, 0, stream>>>(logM, vidx, vlast);
    k_back<<<1, 1024, (TLEN - 1) * 256, stream>>>(vidx, vlast, out + NLOGM);
}